// MyNLLLoss_75136157876603
// MI455X (gfx1250) — compile-verified
//
#include <hip/hip_runtime.h>

// ---- CDNA5 (gfx1250, wave32) weighted-NLL reduction -----------------------
// Stage 1: bandwidth-bound streaming pass, 4 rows/thread via b128 loads with
//          non-temporal hints; wave32 shuffle + LDS reduction -> 1 partial/block.
// Stage 2: single wave32; exact f32 cross-lane reduction on the matrix pipe
//          via V_WMMA_F32_16X16X4_F32 (A = lane partials, B = ones, C = 0).

typedef __attribute__((ext_vector_type(4))) float     v4f;
typedef __attribute__((ext_vector_type(2))) float     v2f;
typedef __attribute__((ext_vector_type(8))) float     v8f;
typedef __attribute__((ext_vector_type(2))) long long v2ll;

#define TPB 256

__global__ void __launch_bounds__(TPB)
nll_partial_kernel(const float* __restrict__ pre,
                   const long long* __restrict__ yt,
                   const float* __restrict__ wptr,
                   float* __restrict__ partial,
                   long long n_rows)
{
    const float w = wptr[0];
    const long long gid      = (long long)blockIdx.x * TPB + threadIdx.x;
    const long long nthreads = (long long)gridDim.x * TPB;
    const long long ngroups  = n_rows >> 2;   // full groups of 4 rows

    float acc = 0.0f;

    for (long long t = gid; t < ngroups; t += nthreads) {
        const long long row = t << 2;
        const v4f*  p4 = (const v4f*)(pre + row * 3);
        const v2ll* y2 = (const v2ll*)(yt + row);

        // 48B of log-probs + 32B of labels, coalesced b128, streaming (NT)
        v4f  f0 = __builtin_nontemporal_load(p4 + 0);
        v4f  f1 = __builtin_nontemporal_load(p4 + 1);
        v4f  f2 = __builtin_nontemporal_load(p4 + 2);
        v2ll ya = __builtin_nontemporal_load(y2 + 0);
        v2ll yb = __builtin_nontemporal_load(y2 + 1);

        const float p[4][3] = {{f0.x, f0.y, f0.z},
                               {f0.w, f1.x, f1.y},
                               {f1.z, f1.w, f2.x},
                               {f2.y, f2.z, f2.w}};
        const int   ys[4]   = {(int)ya.x, (int)ya.y, (int)yb.x, (int)yb.y};

        #pragma unroll
        for (int r = 0; r < 4; ++r) {
            const float p0 = p[r][0], p1 = p[r][1], p2 = p[r][2];
            // argmax, first index wins on ties (matches jnp.argmax)
            int pred = 0; float m = p0;
            if (p1 > m) { m = p1; pred = 1; }
            if (p2 > m) {         pred = 2; }
            const int   yi     = ys[r];
            const float picked = (yi == 0) ? p0 : ((yi == 1) ? p1 : p2);
            const bool  penal  = (pred - yi == 2) || (yi - pred == 2);
            acc += (penal ? w : 1.0f) * picked;
        }
    }

    // scalar tail (N = 2^23 is a multiple of 4; kept for generality)
    if (gid == 0) {
        for (long long row = ngroups << 2; row < n_rows; ++row) {
            const float p0 = pre[row*3+0], p1 = pre[row*3+1], p2 = pre[row*3+2];
            int pred = 0; float m = p0;
            if (p1 > m) { m = p1; pred = 1; }
            if (p2 > m) {         pred = 2; }
            const int   yi     = (int)yt[row];
            const float picked = (yi == 0) ? p0 : ((yi == 1) ? p1 : p2);
            const bool  penal  = (pred - yi == 2) || (yi - pred == 2);
            acc += (penal ? w : 1.0f) * picked;
        }
    }

    // wave32 butterfly reduction (all lanes active here)
    #pragma unroll
    for (int off = 16; off > 0; off >>= 1)
        acc += __shfl_xor(acc, off, 32);

    __shared__ float lds[TPB / 32];
    const int lane = threadIdx.x & 31;
    const int wid  = threadIdx.x >> 5;
    if (lane == 0) lds[wid] = acc;
    __syncthreads();

    if (wid == 0) {                       // whole wave 0 enters: EXEC full
        float v = (lane < TPB / 32) ? lds[lane] : 0.0f;
        #pragma unroll
        for (int off = 4; off > 0; off >>= 1)
            v += __shfl_xor(v, off, 32);
        if (lane == 0) partial[blockIdx.x] = v;
    }
}

// One wave32 block. Reduces per-block partials, then collapses the 32 lane
// sums with V_WMMA_F32_16X16X4_F32 (exact f32 FMA, RNE): B = ones, C = 0 so
// D[m,n] = A[m,0]+A[m,1]+A[m,2]+A[m,3]. Lane m (<16) supplies A[m][0]=v,
// A[m][1]=0; lane m+16 supplies A[m][2]=v, A[m][3]=0. Lane 0 reads row-sums
// 0..7 in d[0..7], lane 16 reads 8..15; one shfl_xor(16) finishes the sum.
__global__ void __launch_bounds__(32)
nll_final_kernel(const float* __restrict__ partial, int nparts,
                 float* __restrict__ out, float inv_n)
{
    float v = 0.0f;
    for (int i = threadIdx.x; i < nparts; i += 32)
        v += partial[i];

    v2f a; a.x = v;    a.y = 0.0f;
    v2f b; b.x = 1.0f; b.y = 1.0f;
    v8f c = {};
    v8f d = __builtin_amdgcn_wmma_f32_16x16x4_f32(
        /*neg_a=*/false, a, /*neg_b=*/false, b,
        /*c_mod=*/(short)0, c, /*reuse_a=*/false, /*reuse_b=*/false);

    float s = d[0] + d[1] + d[2] + d[3] + d[4] + d[5] + d[6] + d[7];
    s += __shfl_xor(s, 16, 32);           // lane0 half + lane16 half

    if (threadIdx.x == 0)
        out[0] = -s * inv_n;
}

extern "C" void kernel_launch(void* const* d_in, const int* in_sizes, int n_in,
                              void* d_out, int out_size, void* d_ws, size_t ws_size,
                              hipStream_t stream)
{
    const float*     pre  = (const float*)d_in[0];      // [N,3] f32 log-softmax
    const long long* yt   = (const long long*)d_in[1];  // [N]   int64
    const float*     wptr = (const float*)d_in[2];      // scalar f32 weight
    float*           out  = (float*)d_out;              // [1]   f32

    const long long n      = (long long)in_sizes[1];    // N rows
    const long long groups = (n + 3) / 4;

    long long blocks = (groups + TPB - 1) / TPB;
    if (blocks > 8192) blocks = 8192;                   // 8192 partials = 32 KB
    const long long wscap = (long long)(ws_size / sizeof(float));
    if (blocks > wscap) blocks = wscap;                 // grid-stride covers rest
    if (blocks < 1) blocks = 1;

    float* partial = (float*)d_ws;

    nll_partial_kernel<<<(int)blocks, TPB, 0, stream>>>(pre, yt, wptr, partial, n);
    nll_final_kernel<<<1, 32, 0, stream>>>(partial, (int)blocks, out,
                                           1.0f / (float)n);
}